// AtlasSpecializedLoss_35167192220158
// MI455X (gfx1250) — compile-verified
//
#include <hip/hip_runtime.h>
#include <hip/hip_bf16.h>
#include <math.h>

// Problem constants (from reference)
#define B_    4096
#define C_    10
#define H_    30
#define W_    30
#define HW_   900
#define NCOL_ 9
#define NPAIR_ 36

// ws float layout:
// 0 focal_sum, 1 exact_count, 2 copy_count, 3 geo_sum, 4 edge_sse, 5 cbal_sse,
// 6 scale_reg_sum, 7 trans_reg_sum, 8 rot_entropy_sum, 9 refl_entropy_sum
typedef float v2f __attribute__((ext_vector_type(2)));
typedef float v8f __attribute__((ext_vector_type(8)));

__global__ void init_ws_kernel(float* ws) {
    if (threadIdx.x < 16) ws[threadIdx.x] = 0.0f;
}

// Wave 0 (lanes 0..31) reduces S[0..255] exactly in f32 using chained
// V_WMMA_F32_16X16X4_F32: D += A(16x4 data) x B(4x16 ones)  -> row sums.
// Total = sum of D rows = (sum of 8 D VGPRs at lane 0) + (same at lane 16).
__device__ __forceinline__ float wave0_wmma_sum256(const float* S, int lane) {
    v2f ones; ones.x = 1.0f; ones.y = 1.0f;
    v8f acc = {};
    #pragma unroll
    for (int k = 0; k < 4; ++k) {
        v2f a;
        a.x = S[k * 64 + lane * 2 + 0];
        a.y = S[k * 64 + lane * 2 + 1];
        acc = __builtin_amdgcn_wmma_f32_16x16x4_f32(
            /*neg_a=*/false, a, /*neg_b=*/false, ones,
            /*c_mod=*/(short)0, acc, /*reuse_a=*/false, /*reuse_b=*/false);
    }
    float s = acc[0] + acc[1] + acc[2] + acc[3] + acc[4] + acc[5] + acc[6] + acc[7];
    return __shfl(s, 0, 32) + __shfl(s, 16, 32);
}

__global__ __launch_bounds__(256)
void loss_main_kernel(const float* __restrict__ pred,
                      const float* __restrict__ target,
                      const float* __restrict__ inputg,
                      float* __restrict__ ws) {
    __shared__ short s_tidx[HW_];
    __shared__ short s_pidx[HW_];
    __shared__ float s_ce[HW_];
    __shared__ float s_pt[HW_];
    __shared__ float s_red[512];
    __shared__ float s_pc[C_];            // softmax mass per class
    __shared__ float s_tc[C_];            // target count per class
    __shared__ float s_cenP[NCOL_ * 3];   // per class 1..9: cnt, sum_y, sum_x (pred idx)
    __shared__ float s_cenT[NCOL_ * 3];   // same for target idx
    __shared__ int   s_exact, s_copy;

    const int tid = threadIdx.x;
    const int b   = blockIdx.x;

    if (tid < C_)        { s_pc[tid] = 0.0f; s_tc[tid] = 0.0f; }
    if (tid < NCOL_ * 3) { s_cenP[tid] = 0.0f; s_cenT[tid] = 0.0f; }
    if (tid == 0)        { s_exact = 1; s_copy = 1; }
    __syncthreads();

    const float* predb = pred   + (size_t)b * C_ * HW_;
    const float* targb = target + (size_t)b * C_ * HW_;
    const float* inpb  = inputg + (size_t)b * C_ * HW_;

    float pcLoc[C_];
    #pragma unroll
    for (int c = 0; c < C_; ++c) pcLoc[c] = 0.0f;
    int exLoc = 1, cpLoc = 1;

    // ---------------- Phase A: per-pixel softmax / argmax / stats -------------
    for (int p = tid; p < HW_; p += 256) {
        float pv[C_];
        #pragma unroll
        for (int c = 0; c < C_; ++c) {
            pv[c] = predb[c * HW_ + p];
            __builtin_prefetch(predb + c * HW_ + p + 256, 0, 1);
        }
        int pidx = 0; float mx = pv[0];
        #pragma unroll
        for (int c = 1; c < C_; ++c) { if (pv[c] > mx) { mx = pv[c]; pidx = c; } }

        // one-hot argmax for target / input; capture pred logit at target class
        int tidx_ = 0, iidx_ = 0; float pv_t = pv[0];
        #pragma unroll
        for (int c = 0; c < C_; ++c) {
            if (targb[c * HW_ + p] > 0.5f) { tidx_ = c; pv_t = pv[c]; }
            if (inpb [c * HW_ + p] > 0.5f) { iidx_ = c; }
        }

        float eArr[C_]; float se = 0.0f, pt_num = 0.0f;
        #pragma unroll
        for (int c = 0; c < C_; ++c) {
            float e = __expf(pv[c] - mx);
            eArr[c] = e; se += e;
            if (c == tidx_) pt_num = e;
        }
        float inv_se = 1.0f / se;
        float ce = (mx + __logf(se)) - pv_t;     // = -log_softmax[target]
        float pt = pt_num * inv_se;              // = exp(-ce)

        s_tidx[p] = (short)tidx_;
        s_pidx[p] = (short)pidx;
        s_ce[p]   = ce;
        s_pt[p]   = pt;

        exLoc &= (pidx == tidx_);
        cpLoc &= (pidx == iidx_);

        #pragma unroll
        for (int c = 0; c < C_; ++c) pcLoc[c] += eArr[c] * inv_se;
        atomicAdd(&s_tc[tidx_], 1.0f);

        const int h = p / W_, w = p % W_;
        if (pidx > 0) {
            atomicAdd(&s_cenP[(pidx - 1) * 3 + 0], 1.0f);
            atomicAdd(&s_cenP[(pidx - 1) * 3 + 1], (float)h);
            atomicAdd(&s_cenP[(pidx - 1) * 3 + 2], (float)w);
        }
        if (tidx_ > 0) {
            atomicAdd(&s_cenT[(tidx_ - 1) * 3 + 0], 1.0f);
            atomicAdd(&s_cenT[(tidx_ - 1) * 3 + 1], (float)h);
            atomicAdd(&s_cenT[(tidx_ - 1) * 3 + 2], (float)w);
        }
    }
    #pragma unroll
    for (int c = 0; c < C_; ++c) atomicAdd(&s_pc[c], pcLoc[c]);
    if (!exLoc) atomicAnd(&s_exact, 0);
    if (!cpLoc) atomicAnd(&s_copy, 0);
    __syncthreads();

    // ---------------- Phase B: edges (focal weight) + Sobel SSE ---------------
    float focalLoc = 0.0f, edgeLoc = 0.0f;
    for (int p = tid; p < HW_; p += 256) {
        const int h = p / W_, w = p % W_;
        const int t0 = s_tidx[p];
        bool e = false;
        if (h > 0      && s_tidx[p - W_] != t0) e = true;
        if (h < H_ - 1 && s_tidx[p + W_] != t0) e = true;
        if (w > 0      && s_tidx[p - 1]  != t0) e = true;
        if (w < W_ - 1 && s_tidx[p + 1]  != t0) e = true;
        const float sw = e ? 1.5f : 1.0f;
        const float ce = s_ce[p];
        const float om = fmaxf(1.0f - s_pt[p], 0.0f);
        focalLoc += __powf(om, 1.4f) * ce * sw;

        // 3x3 Sobel (zero padding), sign-insensitive via magnitude
        auto PV = [&](int hh, int wwi) -> float {
            return (hh >= 0 && hh < H_ && wwi >= 0 && wwi < W_) ? (float)s_pidx[hh * W_ + wwi] : 0.0f;
        };
        auto TV = [&](int hh, int wwi) -> float {
            return (hh >= 0 && hh < H_ && wwi >= 0 && wwi < W_) ? (float)s_tidx[hh * W_ + wwi] : 0.0f;
        };
        float exp_ = -PV(h-1,w-1) + PV(h-1,w+1) - 2.0f*PV(h,w-1) + 2.0f*PV(h,w+1) - PV(h+1,w-1) + PV(h+1,w+1);
        float eyp_ = -PV(h-1,w-1) - 2.0f*PV(h-1,w) - PV(h-1,w+1) + PV(h+1,w-1) + 2.0f*PV(h+1,w) + PV(h+1,w+1);
        float ext_ = -TV(h-1,w-1) + TV(h-1,w+1) - 2.0f*TV(h,w-1) + 2.0f*TV(h,w+1) - TV(h+1,w-1) + TV(h+1,w+1);
        float eyt_ = -TV(h-1,w-1) - 2.0f*TV(h-1,w) - TV(h-1,w+1) + TV(h+1,w-1) + 2.0f*TV(h+1,w) + TV(h+1,w+1);
        float pe = sqrtf(exp_ * exp_ + eyp_ * eyp_);
        float te = sqrtf(ext_ * ext_ + eyt_ * eyt_);
        float d  = pe - te;
        edgeLoc += d * d;
    }
    s_red[tid]       = focalLoc;
    s_red[256 + tid] = edgeLoc;
    __syncthreads();

    // ---------------- Phase C: WMMA reductions + per-image scalars ------------
    if (tid < 32) {   // wave 0 only; EXEC all ones inside
        float focal_tot = wave0_wmma_sum256(&s_red[0],   tid);
        float edge_tot  = wave0_wmma_sum256(&s_red[256], tid);
        if (tid == 0) {
            // geo sample (36 pairs of 9 class centers, compacted over present pairs)
            float cyp[NCOL_], cxp[NCOL_], cyt[NCOL_], cxt[NCOL_];
            bool  prp[NCOL_], prt[NCOL_];
            for (int k = 0; k < NCOL_; ++k) {
                float cp = s_cenP[k * 3 + 0];
                prp[k] = cp > 0.0f;
                cyp[k] = s_cenP[k * 3 + 1] / fmaxf(cp, 1.0f);
                cxp[k] = s_cenP[k * 3 + 2] / fmaxf(cp, 1.0f);
                float ct = s_cenT[k * 3 + 0];
                prt[k] = ct > 0.0f;
                cyt[k] = s_cenT[k * 3 + 1] / fmaxf(ct, 1.0f);
                cxt[k] = s_cenT[k * 3 + 2] / fmaxf(ct, 1.0f);
            }
            float dpc[NPAIR_], dtc[NPAIR_];
            int np_ = 0, nt_ = 0;
            for (int i = 0; i < NCOL_; ++i) {
                for (int j = i + 1; j < NCOL_; ++j) {
                    if (prp[i] && prp[j]) {
                        float dy = cyp[i] - cyp[j], dx = cxp[i] - cxp[j];
                        dpc[np_++] = sqrtf(dy * dy + dx * dx);
                    }
                    if (prt[i] && prt[j]) {
                        float dy = cyt[i] - cyt[j], dx = cxt[i] - cxt[j];
                        dtc[nt_++] = sqrtf(dy * dy + dx * dx);
                    }
                }
            }
            int mm = np_ < nt_ ? np_ : nt_;
            float gs = 0.0f;
            for (int k = 0; k < mm; ++k) { float d = dpc[k] - dtc[k]; gs += d * d; }
            float geo_b = (mm > 0) ? gs / (float)mm : 0.0f;

            // class balance SSE for this image
            float ps = 0.0f, ts = 0.0f;
            for (int c = 0; c < C_; ++c) { ps += s_pc[c]; ts += s_tc[c]; }
            float ip = 1.0f / (ps + 1e-8f), it = 1.0f / (ts + 1e-8f);
            float cb = 0.0f;
            for (int c = 0; c < C_; ++c) {
                float d = s_pc[c] * ip - s_tc[c] * it;
                cb += d * d;
            }

            atomicAdd(&ws[0], focal_tot);
            atomicAdd(&ws[1], (float)s_exact);
            atomicAdd(&ws[2], (float)s_copy);
            atomicAdd(&ws[3], geo_b);
            atomicAdd(&ws[4], edge_tot);
            atomicAdd(&ws[5], cb);
        }
    }
}

__device__ __forceinline__ float entropy_n(const float* x, int n) {
    float m = x[0];
    for (int i = 1; i < n; ++i) m = fmaxf(m, x[i]);
    float se = 0.0f;
    for (int i = 0; i < n; ++i) se += __expf(x[i] - m);
    float lse = m + __logf(se);
    float H = 0.0f;
    for (int i = 0; i < n; ++i) {
        float p = __expf(x[i] - lse);
        H += p * (lse - x[i]);
    }
    return H;
}

__global__ __launch_bounds__(256)
void small_inputs_kernel(const float* __restrict__ theta,
                         const float* __restrict__ rot,
                         const float* __restrict__ refl,
                         float* __restrict__ ws) {
    int b = blockIdx.x * blockDim.x + threadIdx.x;
    float scale = 0.0f, trans = 0.0f, hr = 0.0f, hf = 0.0f;
    if (b < B_) {
        const float* t = theta + (size_t)b * 6;
        scale = sqrtf(t[0] * t[0] + t[1] * t[1] + t[3] * t[3] + t[4] * t[4]);
        trans = sqrtf(t[2] * t[2] + t[5] * t[5]);
        float r8[8];
        for (int i = 0; i < 8; ++i) r8[i] = rot[(size_t)b * 8 + i];
        hr = entropy_n(r8, 8);
        float r4[4];
        for (int i = 0; i < 4; ++i) r4[i] = refl[(size_t)b * 4 + i];
        hf = entropy_n(r4, 4);
    }
    atomicAdd(&ws[6], scale);
    atomicAdd(&ws[7], trans);
    atomicAdd(&ws[8], hr);
    atomicAdd(&ws[9], hf);
}

__global__ void finalize_kernel(const float* __restrict__ ws, float* __restrict__ out) {
    if (threadIdx.x == 0 && blockIdx.x == 0) {
        const float invN = 1.0f / (float)(B_ * HW_);
        const float invB = 1.0f / (float)B_;
        float spatial_focal = ws[0] * invN;
        float exact_count   = ws[1];
        float exact_bonus   = -(ws[1] * invB) * 7.0f;
        float transform     = ws[2] * invB * 0.2f;
        float geo           = ws[3] * invB * 0.5f;
        float edge          = ws[4] * invN * 0.3f;
        float cbal          = ws[5] * invB * (1.0f / (float)C_) * 0.2f;
        float affine        = (ws[6] * invB + 0.1f * ws[7] * invB) * 0.4f;
        float rotation      = ws[8] * invB * 0.3f;
        float reflection    = ws[9] * invB * 0.3f;
        float total = spatial_focal + transform + affine + rotation + reflection +
                      geo + edge + cbal + exact_bonus;
        out[0]  = total;
        out[1]  = spatial_focal;
        out[2]  = transform;
        out[3]  = exact_bonus;
        out[4]  = exact_count;
        out[5]  = affine;
        out[6]  = rotation;
        out[7]  = reflection;
        out[8]  = geo;
        out[9]  = edge;
        out[10] = cbal;
    }
}

extern "C" void kernel_launch(void* const* d_in, const int* in_sizes, int n_in,
                              void* d_out, int out_size, void* d_ws, size_t ws_size,
                              hipStream_t stream) {
    const float* pred   = (const float*)d_in[0];
    const float* target = (const float*)d_in[1];
    const float* inputg = (const float*)d_in[2];
    const float* theta  = (const float*)d_in[3];
    const float* rot    = (const float*)d_in[4];
    const float* refl   = (const float*)d_in[5];
    float* ws  = (float*)d_ws;
    float* out = (float*)d_out;

    hipLaunchKernelGGL(init_ws_kernel, dim3(1), dim3(32), 0, stream, ws);
    hipLaunchKernelGGL(loss_main_kernel, dim3(B_), dim3(256), 0, stream,
                       pred, target, inputg, ws);
    hipLaunchKernelGGL(small_inputs_kernel, dim3(B_ / 256), dim3(256), 0, stream,
                       theta, rot, refl, ws);
    hipLaunchKernelGGL(finalize_kernel, dim3(1), dim3(1), 0, stream, ws, out);
}